// RotationGate_32186484916946
// MI455X (gfx1250) — compile-verified
//
#include <hip/hip_runtime.h>
#include <cstdint>

// RX gate on qubit 10 of a 24-qubit state, batch=2, fp32 re/im planes.
// Pure streaming kernel (AI ~= 0.25 flop/byte): bandwidth-bound, ~22us floor
// at 23.3 TB/s for 256 MB in + 256 MB out. WMMA deliberately NOT used: the
// 2x2 gate needs < 6 TFLOP/s of VALU to saturate HBM, and WMMA lane layouts
// would break the coalesced 128-bit stream.
// CDNA5 path: double-buffered global_load_async_to_lds_b128 (ASYNCcnt),
// ds_load_b128 consume, nontemporal global_store_b128 output.

typedef __attribute__((ext_vector_type(4))) float v4f;

constexpr uint32_t kNQ         = 24;
constexpr uint32_t kTarget     = 10;
constexpr uint32_t kDim        = 1u << kNQ;                 // 2^24
constexpr uint32_t kBatch      = 2;
constexpr uint32_t kRight      = 1u << (kNQ - kTarget - 1); // 8192
constexpr uint32_t kInnerF     = kRight * kBatch;           // 16384 floats between pair partners
constexpr uint32_t kInnerV4    = kInnerF / 4;               // 4096 float4 per 'a' block (2^12)
constexpr uint32_t kItems      = (kDim * kBatch) / 2 / 4;   // 2^22 float4-pair work items
constexpr uint32_t kPartStride = kDim * kBatch;             // 2^25 floats between re/im output parts

#define THREADS 256
#define BLOCKS  4096   // kItems / (BLOCKS*THREADS) == 4 iterations, no tail

__global__ __launch_bounds__(THREADS) void rx_gate_stream_kernel(
    const float* __restrict__ sre, const float* __restrict__ sim,
    const float* __restrict__ thetas, float* __restrict__ out)
{
    // 2 stages x {re0,re1,im0,im1} x 256 lanes x 16B = 32 KB. Per-wave-private
    // slots: each wave reads only LDS written by its OWN async loads, so the
    // only ordering needed is s_wait_asynccnt (no workgroup barriers).
    __shared__ v4f stage[2][4][THREADS];

    const uint32_t tid    = threadIdx.x;
    const uint32_t gtid   = blockIdx.x * THREADS + tid;
    const uint32_t stride = gridDim.x * THREADS;

    // Per-batch rotation coefficients. thetas ~ N(0,1) so |theta/2| is tiny:
    // use the HW transcendentals (v_sin_f32/v_cos_f32) instead of libm's
    // Payne-Hanek path, which emitted a ~700-instruction scalar blob.
    const float t0 = 0.5f * thetas[0];
    const float t1 = 0.5f * thetas[1];
    const float c0 = __cosf(t0);
    const float s0 = __sinf(t0);
    const float c1 = __cosf(t1);
    const float s1 = __sinf(t1);
    // float4 components 0,2 are batch 0; components 1,3 are batch 1.
    const v4f C = {c0, c1, c0, c1};
    const v4f S = {s0, s1, s0, s1};

    // Issue 4 async 128-bit global->LDS copies for work item u4 into buffer `buf`.
    auto issue = [&](int buf, uint32_t u4) {
        const uint32_t a     = u4 >> 12;            // / kInnerV4
        const uint32_t inner = u4 & (kInnerV4 - 1);
        const uint32_t f0    = a * (2u * kInnerF) + inner * 4u;
        const uint32_t f1    = f0 + kInnerF;
        const float* p0 = sre + f0;   // re, target bit 0
        const float* p1 = sre + f1;   // re, target bit 1
        const float* p2 = sim + f0;   // im, target bit 0
        const float* p3 = sim + f1;   // im, target bit 1
        // Generic pointers to __shared__ carry the LDS byte offset in addr[31:0].
        const uint32_t l0 = (uint32_t)(uintptr_t)&stage[buf][0][tid];
        const uint32_t l1 = (uint32_t)(uintptr_t)&stage[buf][1][tid];
        const uint32_t l2 = (uint32_t)(uintptr_t)&stage[buf][2][tid];
        const uint32_t l3 = (uint32_t)(uintptr_t)&stage[buf][3][tid];
        asm volatile(
            "global_load_async_to_lds_b128 %0, %4, off\n\t"
            "global_load_async_to_lds_b128 %1, %5, off\n\t"
            "global_load_async_to_lds_b128 %2, %6, off\n\t"
            "global_load_async_to_lds_b128 %3, %7, off"
            :
            : "v"(l0), "v"(l1), "v"(l2), "v"(l3),
              "v"(p0), "v"(p1), "v"(p2), "v"(p3)
            : "memory");
    };

    uint32_t i = gtid;
    if (i < kItems) issue(0, i);     // prime stage 0
    int buf = 0;

    for (; i < kItems; i += stride) {
        const uint32_t nxt = i + stride;        // wave-uniform predicate (exact grid fit)
        if (nxt < kItems) {
            issue(buf ^ 1, nxt);                // prefetch next stage first
            // Leave only the 4 just-issued copies outstanding: the current
            // stage's 4 copies (issued earlier, done-in-order) are complete.
            asm volatile("s_wait_asynccnt 0x4" ::: "memory");
        } else {
            asm volatile("s_wait_asynccnt 0x0" ::: "memory");
        }

        const v4f re0 = stage[buf][0][tid];     // ds_load_b128
        const v4f re1 = stage[buf][1][tid];
        const v4f im0 = stage[buf][2][tid];
        const v4f im1 = stage[buf][3][tid];

        // U = cos(t/2) I - i sin(t/2) X applied to (re,im) pairs.
        const v4f ore0 = C * re0 + S * im1;
        const v4f ore1 = C * re1 + S * im0;
        const v4f oim0 = C * im0 - S * re1;
        const v4f oim1 = C * im1 - S * re0;

        const uint32_t a     = i >> 12;
        const uint32_t inner = i & (kInnerV4 - 1);
        const uint32_t f0    = a * (2u * kInnerF) + inner * 4u;
        const uint32_t f1    = f0 + kInnerF;

        // Output written once, never re-read: non-temporal 128-bit stores.
        __builtin_nontemporal_store(ore0, (v4f*)(out + f0));
        __builtin_nontemporal_store(ore1, (v4f*)(out + f1));
        __builtin_nontemporal_store(oim0, (v4f*)(out + kPartStride + f0));
        __builtin_nontemporal_store(oim1, (v4f*)(out + kPartStride + f1));

        buf ^= 1;
    }
}

extern "C" void kernel_launch(void* const* d_in, const int* in_sizes, int n_in,
                              void* d_out, int out_size, void* d_ws, size_t ws_size,
                              hipStream_t stream) {
    const float* state_re = (const float*)d_in[0];
    const float* state_im = (const float*)d_in[1];
    const float* thetas   = (const float*)d_in[2];
    float*       out      = (float*)d_out;

    rx_gate_stream_kernel<<<BLOCKS, THREADS, 0, stream>>>(state_re, state_im, thetas, out);
}